// QuanvolutionSamplerHybrid_65481071407064
// MI455X (gfx1250) — compile-verified
//
#include <hip/hip_runtime.h>
#include <hip/hip_bf16.h>

// WMMA vector types (wave32, gfx1250)
typedef __attribute__((ext_vector_type(2))) float v2f;
typedef __attribute__((ext_vector_type(8))) float v8f;

#define KSPLIT       7
#define PATCHES_PER  28              // 196 / 7
#define LK_PER       (PATCHES_PER*20)  // 560 local K values per range
#define WSTRIDE      562             // padded LDS row stride: 50*m mod 64 distinct for m=0..15

__global__ void zero_ws_kernel(float* ws, int n) {
    int i = blockIdx.x * blockDim.x + threadIdx.x;
    if (i < n) ws[i] = 0.0f;
}

// One block = 256 threads = 8 waves. Each wave: 16 images x 16 classes tile,
// accumulating over 28 patches (20 features each) via V_WMMA_F32_16X16X4_F32.
__global__ __launch_bounds__(256) void quanv_wmma_kernel(
    const float* __restrict__ x,      // (nImg, 784)
    const float* __restrict__ fc_w,   // (10, 3920)
    float*       __restrict__ logits, // (nImg, 16) partial accumulators
    int nImg)
{
    __shared__ float wlds[16 * WSTRIDE];

    const int tid      = threadIdx.x;
    const int krange   = blockIdx.y;
    const int patch_lo = krange * PATCHES_PER;

    // Stage weights for this patch range into LDS in WMMA-A-friendly order:
    // wlds[m][lp*20 + j]  (j<16: filter feature j of patch, j>=16: sampler j-16)
    for (int idx = tid; idx < 16 * LK_PER; idx += 256) {
        int m     = idx / LK_PER;
        int lk    = idx % LK_PER;
        int patch = patch_lo + lk / 20;
        int j     = lk % 20;
        int gk    = (j < 16) ? (patch * 16 + j) : (3136 + patch * 4 + (j - 16));
        float w   = (m < 10) ? fc_w[m * 3920 + gk] : 0.0f;
        wlds[m * WSTRIDE + lk] = w;
    }
    __syncthreads();

    const int wave = tid >> 5;
    const int lane = tid & 31;
    const int half = lane >> 4;      // 0: K0/K1 half, 1: K2/K3 half
    const int mrow = lane & 15;      // class row (A) / image column (B,D)
    const int img  = blockIdx.x * 128 + wave * 16 + mrow;
    const float* xi = x + (size_t)img * 784;

    v8f acc = {0.0f, 0.0f, 0.0f, 0.0f, 0.0f, 0.0f, 0.0f, 0.0f};

    for (int lp = 0; lp < PATCHES_PER; ++lp) {
        const int patch = patch_lo + lp;
        const int pi = patch / 14, pj = patch % 14;
        const int off = pi * 56 + pj * 2;            // row 2*pi, col 2*pj
        // 8-byte aligned pixel pair loads: (p0,p1) and (p2,p3)
        const float2 top = *(const float2*)(xi + off);
        const float2 bot = *(const float2*)(xi + off + 28);
        const float p0 = top.x, p1 = top.y, p2 = bot.x, p3 = bot.y;

        const float c0 = __cosf(0.5f * p0), s0 = __sinf(0.5f * p0);
        const float c1 = __cosf(0.5f * p1), s1 = __sinf(0.5f * p1);
        const float c2 = __cosf(0.5f * p2), s2 = __sinf(0.5f * p2);
        const float c3 = __cosf(0.5f * p3), s3 = __sinf(0.5f * p3);
        (void)c2; (void)c3;

        // 2-qubit amplitude grid a[q0][q1], 8x [CX(0,1); RY(p0) on q0]
        float a00 = c0 * c1, a01 = c0 * s1, a10 = s0 * c1, a11 = s0 * s1;
        #pragma unroll
        for (int it = 0; it < 8; ++it) {
            const float t10 = a11, t11 = a10;        // CX: swap row q0=1
            const float n00 = c0 * a00 - s0 * t10;   // RY(p0) on q0
            const float n01 = c0 * a01 - s0 * t11;
            const float n10 = s0 * a00 + c0 * t10;
            const float n11 = s0 * a01 + c0 * t11;
            a00 = n00; a01 = n01; a10 = n10; a11 = n11;
        }
        const float P00 = a00 * a00, P01 = a01 * a01;
        const float P10 = a10 * a10, P11 = a11 * a11;
        const float p2s = s2 * s2, p3s = s3 * s3;
        const float q2a = 1.0f - p2s, q3a = 1.0f - p3s;

        float f[20];
        f[ 0] = P00 * q2a * q3a;  f[ 1] = P00 * q2a * p3s;
        f[ 2] = P00 * p2s * q3a;  f[ 3] = P00 * p2s * p3s;
        f[ 4] = P01 * q2a * q3a;  f[ 5] = P01 * q2a * p3s;
        f[ 6] = P01 * p2s * q3a;  f[ 7] = P01 * p2s * p3s;
        f[ 8] = P10 * q2a * q3a;  f[ 9] = P10 * q2a * p3s;
        f[10] = P10 * p2s * q3a;  f[11] = P10 * p2s * p3s;
        f[12] = P11 * q2a * q3a;  f[13] = P11 * q2a * p3s;
        f[14] = P11 * p2s * q3a;  f[15] = P11 * p2s * p3s;
        const float ss0 = s0 * s0, ss3 = s3 * s3;    // sampler: sin(p0/2)^2, sin(p3/2)^2
        f[16] = (1.0f - ss0) * (1.0f - ss3);
        f[17] = (1.0f - ss0) * ss3;
        f[18] = ss0 * (1.0f - ss3);
        f[19] = ss0 * ss3;

        // 5 WMMA steps of K=4. A (weights) from LDS in native 16x4 layout:
        // lanes 0-15 get K0/K1, lanes 16-31 get K2/K3 (8B-aligned ds_load_b64).
        // B (features): this lane's half-wave natively owns the right K pair.
        const int abase = mrow * WSTRIDE + lp * 20 + half * 2;
        #pragma unroll
        for (int t = 0; t < 5; ++t) {
            const float2 aw = *(const float2*)&wlds[abase + 4 * t];
            v2f a; a.x = aw.x; a.y = aw.y;
            v2f b;
            b.x = half ? f[4 * t + 2] : f[4 * t + 0];
            b.y = half ? f[4 * t + 3] : f[4 * t + 1];
            acc = __builtin_amdgcn_wmma_f32_16x16x4_f32(
                false, a, false, b, (short)0, acc, false, false);
        }
    }

    // D layout: lane column N = mrow (image), rows M = r + half*8.
    union { v8f v; float s[8]; } u; u.v = acc;
    #pragma unroll
    for (int r = 0; r < 8; ++r) {
        const int M = r + half * 8;                  // classes 10..15 are junk but unread
        atomicAdd(&logits[(size_t)img * 16 + M], u.s[r]);
    }
}

__global__ void logsoftmax_kernel(const float* __restrict__ logits,
                                  const float* __restrict__ fc_b,
                                  float* __restrict__ out, int nImg)
{
    int img = blockIdx.x * blockDim.x + threadIdx.x;
    if (img >= nImg) return;
    float l[10];
    float mx = -1e30f;
    #pragma unroll
    for (int m = 0; m < 10; ++m) {
        l[m] = logits[(size_t)img * 16 + m] + fc_b[m];
        mx = fmaxf(mx, l[m]);
    }
    float s = 0.0f;
    #pragma unroll
    for (int m = 0; m < 10; ++m) s += __expf(l[m] - mx);
    const float lz = mx + __logf(s);
    #pragma unroll
    for (int m = 0; m < 10; ++m) out[(size_t)img * 10 + m] = l[m] - lz;
}

extern "C" void kernel_launch(void* const* d_in, const int* in_sizes, int n_in,
                              void* d_out, int out_size, void* d_ws, size_t ws_size,
                              hipStream_t stream) {
    const float* x    = (const float*)d_in[0];   // (B,1,28,28) f32
    const float* fc_w = (const float*)d_in[1];   // (10,3920)  f32
    const float* fc_b = (const float*)d_in[2];   // (10,)      f32
    float* out = (float*)d_out;                  // (B,10)     f32
    const int nImg = in_sizes[0] / 784;          // 8192

    float* logits = (float*)d_ws;                // (nImg,16) partials, 512 KB
    const int nz = nImg * 16;
    zero_ws_kernel<<<(nz + 255) / 256, 256, 0, stream>>>(logits, nz);

    dim3 grid(nImg / 128, KSPLIT);               // 64 image-tiles x 7 K-ranges
    quanv_wmma_kernel<<<grid, 256, 0, stream>>>(x, fc_w, logits, nImg);

    logsoftmax_kernel<<<(nImg + 255) / 256, 256, 0, stream>>>(logits, fc_b, out, nImg);
}